// TensorProduct_70480413327405
// MI455X (gfx1250) — compile-verified
//
#include <hip/hip_runtime.h>

typedef __attribute__((ext_vector_type(2))) float v2f;
typedef __attribute__((ext_vector_type(8))) float v8f;

#define SH_STRIDE 292   // 288 + 4 pad: bank-stride 36 -> conflict-free over 16 rows

// ---------------- Path tables: sparse Wigner-3j entries, alpha pre-folded -----------
// alpha(lo=0)=sqrt(1/3072), alpha(lo=1)=sqrt(3/4096), alpha(lo=2)=sqrt(5/4096)
// l=1 basis order (y,z,x); l=2 basis order (xy,yz,z2,zx,x2-y2) [e3nn m=-l..l]

struct P0 { static constexpr int IDX=0,L1=0,L2=0,LO=0,OFF1=0,OFF2=0,NE=1;
  static constexpr int I[1]={0},J[1]={0},K[1]={0};
  static constexpr float V[1]={0.01804220f}; };

struct P1 { static constexpr int IDX=1,L1=1,L2=1,LO=0,OFF1=32,OFF2=32,NE=3;
  static constexpr int I[3]={0,1,2},J[3]={0,1,2},K[3]={0,0,0};
  static constexpr float V[3]={0.01041667f,0.01041667f,0.01041667f}; };

struct P2 { static constexpr int IDX=2,L1=2,L2=2,LO=0,OFF1=128,OFF2=128,NE=5;
  static constexpr int I[5]={0,1,2,3,4},J[5]={0,1,2,3,4},K[5]={0,0,0,0,0};
  static constexpr float V[5]={0.00806872f,0.00806872f,0.00806872f,0.00806872f,0.00806872f}; };

struct P3 { static constexpr int IDX=3,L1=0,L2=1,LO=1,OFF1=0,OFF2=32,NE=3;
  static constexpr int I[3]={0,0,0},J[3]={0,1,2},K[3]={0,1,2};
  static constexpr float V[3]={0.015625f,0.015625f,0.015625f}; };

struct P4 { static constexpr int IDX=4,L1=1,L2=0,LO=1,OFF1=32,OFF2=0,NE=3;
  static constexpr int I[3]={0,1,2},J[3]={0,0,0},K[3]={0,1,2};
  static constexpr float V[3]={0.015625f,0.015625f,0.015625f}; };

// C(1,2,1)[i,j,k] = C(1,1,2)[i,k,j] * alpha1
struct P5 { static constexpr int IDX=5,L1=1,L2=2,LO=1,OFF1=32,OFF2=128,NE=11;
  static constexpr int I[11]={2,0,0,1,0,1,2,1,2,0,2},
                      J[11]={0,0,1,1,2,2,2,3,3,4,4},
                      K[11]={0,2,1,0,0,1,2,2,1,0,2};
  static constexpr float V[11]={ 0.00855816f, 0.00855816f, 0.00855816f, 0.00855816f,
                                -0.00494103f, 0.00988206f,-0.00494103f, 0.00855816f,
                                 0.00855816f,-0.00855816f, 0.00855816f}; };

// C(2,1,1)[i,j,k] = C(1,1,2)[j,k,i] * alpha1
struct P6 { static constexpr int IDX=6,L1=2,L2=1,LO=1,OFF1=128,OFF2=32,NE=11;
  static constexpr int I[11]={0,0,1,1,2,2,2,3,3,4,4},
                      J[11]={2,0,0,1,0,1,2,1,2,0,2},
                      K[11]={0,2,1,0,0,1,2,2,1,0,2};
  static constexpr float V[11]={ 0.00855816f, 0.00855816f, 0.00855816f, 0.00855816f,
                                -0.00494103f, 0.00988206f,-0.00494103f, 0.00855816f,
                                 0.00855816f,-0.00855816f, 0.00855816f}; };

struct P7 { static constexpr int IDX=7,L1=0,L2=2,LO=2,OFF1=0,OFF2=128,NE=5;
  static constexpr int I[5]={0,0,0,0,0},J[5]={0,1,2,3,4},K[5]={0,1,2,3,4};
  static constexpr float V[5]={0.015625f,0.015625f,0.015625f,0.015625f,0.015625f}; };

// C(1,1,2) * alpha2
struct P8 { static constexpr int IDX=8,L1=1,L2=1,LO=2,OFF1=32,OFF2=32,NE=11;
  static constexpr int I[11]={2,0,0,1,0,1,2,1,2,0,2},
                      J[11]={0,2,1,0,0,1,2,2,1,0,2},
                      K[11]={0,0,1,1,2,2,2,3,3,4,4};
  static constexpr float V[11]={ 0.01104854f, 0.01104854f, 0.01104854f, 0.01104854f,
                                -0.00637887f, 0.01275775f,-0.00637887f, 0.01104854f,
                                 0.01104854f,-0.01104854f, 0.01104854f}; };

struct P9 { static constexpr int IDX=9,L1=2,L2=0,LO=2,OFF1=128,OFF2=0,NE=5;
  static constexpr int I[5]={0,1,2,3,4},J[5]={0,0,0,0,0},K[5]={0,1,2,3,4};
  static constexpr float V[5]={0.015625f,0.015625f,0.015625f,0.015625f,0.015625f}; };

// C(2,2,2): fully symmetric Tr(Ma Mb Mc) invariants * alpha2 (25 entries)
struct P10 { static constexpr int IDX=10,L1=2,L2=2,LO=2,OFF1=128,OFF2=128,NE=25;
  static constexpr int I[25]={2, 2,0,0, 2,4,4, 2,1,1, 2,3,3, 4,1,1, 4,3,3, 0,0,1,1,3,3},
                      J[25]={2, 0,2,0, 4,2,4, 1,2,1, 3,2,3, 1,4,1, 3,4,3, 1,3,0,3,0,1},
                      K[25]={2, 0,0,2, 4,4,2, 1,1,2, 3,3,2, 1,1,4, 3,3,4, 3,1,3,0,1,0};
  static constexpr float V[25]={ 0.00835185f,
    -0.00835185f,-0.00835185f,-0.00835185f,
    -0.00835185f,-0.00835185f,-0.00835185f,
     0.00417593f, 0.00417593f, 0.00417593f,
     0.00417593f, 0.00417593f, 0.00417593f,
    -0.00723286f,-0.00723286f,-0.00723286f,
     0.00723286f, 0.00723286f, 0.00723286f,
     0.00723286f, 0.00723286f, 0.00723286f, 0.00723286f, 0.00723286f, 0.00723286f}; };

// ---------------- per-path WMMA loop -------------------------------------------------
// D[(z),(w)] += sum_uv A[z,uv] * B[uv,w], K streamed 4 uv at a time with
// v_wmma_f32_16x16x4_f32; A built on the fly from sparse CG * s1 * s2 in LDS.
template<typename PT>
__device__ __forceinline__ void run_path(const float* __restrict__ ws,
                                         const float* __restrict__ sh1,
                                         const float* __restrict__ sh2,
                                         int lane, int wh, v8f* acc)
{
    constexpr int D1 = 2*PT::L1 + 1;
    constexpr int D2 = 2*PT::L2 + 1;
    constexpr int DO = 2*PT::LO + 1;
    const int z = lane & 15;
    const int h = lane >> 4;
    const int n = lane & 15;
    const float* __restrict__ r1 = sh1 + z*SH_STRIDE + PT::OFF1;
    const float* __restrict__ r2 = sh2 + z*SH_STRIDE + PT::OFF2;
    // B base: ws[p][uv][w] flat p*32768 + uv*32 + w ; this lane covers w = wh*16+n,
    // K rows (uv + 2h) and (uv + 2h + 1)
    const float* __restrict__ wb = ws + (size_t)PT::IDX*32768 + wh*16 + n + 2*h*32;
    __builtin_prefetch(wb, 0, 1);   // global_prefetch_b8 (ws is L2 resident)

    for (int u = 0; u < 32; ++u) {
        float s1v[D1];
#pragma unroll
        for (int i = 0; i < D1; ++i) s1v[i] = r1[u*D1 + i];
        const float* __restrict__ wbu = wb + u*1024;

        for (int vb = 0; vb < 32; vb += 4) {
            float t0[DO], t1[DO];
#pragma unroll
            for (int k = 0; k < DO; ++k) { t0[k] = 0.f; t1[k] = 0.f; }
            const int v0 = vb + 2*h;
            {
                float s2v[D2];
#pragma unroll
                for (int j = 0; j < D2; ++j) s2v[j] = r2[v0*D2 + j];
#pragma unroll
                for (int e = 0; e < PT::NE; ++e)
                    t0[PT::K[e]] += PT::V[e] * s1v[PT::I[e]] * s2v[PT::J[e]];
            }
            {
                float s2v[D2];
#pragma unroll
                for (int j = 0; j < D2; ++j) s2v[j] = r2[(v0+1)*D2 + j];
#pragma unroll
                for (int e = 0; e < PT::NE; ++e)
                    t1[PT::K[e]] += PT::V[e] * s1v[PT::I[e]] * s2v[PT::J[e]];
            }
            v2f b;
            b.x = wbu[vb*32];        // K row uv+2h
            b.y = wbu[vb*32 + 32];   // K row uv+2h+1
#pragma unroll
            for (int k = 0; k < DO; ++k) {
                v2f a; a.x = t0[k]; a.y = t1[k];
                acc[k] = __builtin_amdgcn_wmma_f32_16x16x4_f32(
                    false, a, false, b, (short)0, acc[k], false, false);
            }
        }
    }
}

template<int DO, int OFFO>
__device__ __forceinline__ void store_out(float* __restrict__ out, int zt,
                                          int lane, int wh, const v8f* acc)
{
    const int h = lane >> 4, n = lane & 15;
#pragma unroll
    for (int r = 0; r < 8; ++r) {
        const int z = r + 8*h;                       // C/D layout: VGPR r -> M = r (+8 for hi lanes)
        float* __restrict__ po = out + (size_t)(zt*16 + z)*288 + OFFO + (wh*16 + n)*DO;
#pragma unroll
        for (int k = 0; k < DO; ++k) po[k] = acc[k][r];
    }
}

__global__ __launch_bounds__(64)
void tp_kernel(const float* __restrict__ x1, const float* __restrict__ x2,
               const float* __restrict__ ws, float* __restrict__ out)
{
    __shared__ __align__(16) float sh1[16*SH_STRIDE];
    __shared__ __align__(16) float sh2[16*SH_STRIDE];
    const int zt  = blockIdx.x;          // 16-row batch tile
    const int io  = blockIdx.y;          // output irrep 0/1/2
    const int tid = threadIdx.x;

    const float* __restrict__ px1 = x1 + (size_t)zt*16*288;
    const float* __restrict__ px2 = x2 + (size_t)zt*16*288;
    for (int idx = tid; idx < 16*72; idx += 64) {        // 72 float4 per 288-f row
        const int row = idx / 72, c = idx - row*72;
        ((float4*)(sh1 + row*SH_STRIDE))[c] = ((const float4*)(px1 + (size_t)row*288))[c];
        ((float4*)(sh2 + row*SH_STRIDE))[c] = ((const float4*)(px2 + (size_t)row*288))[c];
    }
    __syncthreads();

    const int lane = tid & 31;
    const int wh   = tid >> 5;           // wave 0: w 0..15, wave 1: w 16..31

    if (io == 0) {
        v8f acc[1] = {};
        run_path<P0>(ws, sh1, sh2, lane, wh, acc);
        run_path<P1>(ws, sh1, sh2, lane, wh, acc);
        run_path<P2>(ws, sh1, sh2, lane, wh, acc);
        store_out<1, 0>(out, zt, lane, wh, acc);
    } else if (io == 1) {
        v8f acc[3] = {};
        run_path<P3>(ws, sh1, sh2, lane, wh, acc);
        run_path<P4>(ws, sh1, sh2, lane, wh, acc);
        run_path<P5>(ws, sh1, sh2, lane, wh, acc);
        run_path<P6>(ws, sh1, sh2, lane, wh, acc);
        store_out<3, 32>(out, zt, lane, wh, acc);
    } else {
        v8f acc[5] = {};
        run_path<P7>(ws, sh1, sh2, lane, wh, acc);
        run_path<P8>(ws, sh1, sh2, lane, wh, acc);
        run_path<P9>(ws, sh1, sh2, lane, wh, acc);
        run_path<P10>(ws, sh1, sh2, lane, wh, acc);
        store_out<5, 128>(out, zt, lane, wh, acc);
    }
}

extern "C" void kernel_launch(void* const* d_in, const int* in_sizes, int n_in,
                              void* d_out, int out_size, void* d_ws, size_t ws_size,
                              hipStream_t stream)
{
    const float* x1 = (const float*)d_in[0];
    const float* x2 = (const float*)d_in[1];
    const float* ws = (const float*)d_in[2];
    float* out = (float*)d_out;
    const int B = in_sizes[0] / 288;     // 2048
    dim3 grid(B / 16, 3), block(64);
    hipLaunchKernelGGL(tp_kernel, grid, block, 0, stream, x1, x2, ws, out);
}